// GraphSAGE_4784593568513
// MI455X (gfx1250) — compile-verified
//
#include <hip/hip_runtime.h>
#include <stdint.h>

// ---------------------------------------------------------------------------
// GraphSAGE (max aggregation), 2 layers, d=64, fused for MI455X (gfx1250).
//   agg[i] = max_{e: dst[e]==i} x[src[e]]   (empty -> 0)
//   h      = relu(agg @ Wl1 + x @ Wr1 + b1)
//   out    = agg2 @ Wl2 + h @ Wr2 + b2     (d_out = 1)
// Aggregation: native global_atomic_max_u32 on monotonic-encoded f32.
// Dense lin layers: V_WMMA_F32_16X16X4_F32 (full fp32 precision).
// ---------------------------------------------------------------------------

typedef __attribute__((ext_vector_type(2))) float v2f;
typedef __attribute__((ext_vector_type(8))) float v8f;

#define D   64      // feature width (d_in == d_h == 64)
#define LDA 68      // padded LDS row stride for A tiles (banks: 68 % 64 != 0)

// Order-preserving f32 -> u32 encoding so unsigned atomicMax == float max.
__device__ __forceinline__ unsigned f2mono(float f) {
    unsigned u = __float_as_uint(f);
    return (u & 0x80000000u) ? ~u : (u | 0x80000000u);
}
// Decode; encoded 0 is the init sentinel (= "no in-edges") -> 0.0f like PyG.
__device__ __forceinline__ float mono2f_or_zero(unsigned e) {
    if (e == 0u) return 0.0f;
    return __uint_as_float((e & 0x80000000u) ? (e & 0x7fffffffu) : ~e);
}

// ---------------------------------------------------------------------------
__global__ void k_init_enc(unsigned* __restrict__ agg, int n) {
    int i = blockIdx.x * blockDim.x + threadIdx.x;
    if (i < n) agg[i] = 0u;
}

// One edge handled by 16 threads (float4 feature chunks): coalesced 256B row
// gather, then 4 native u32 max-atomics into the destination row.
__global__ void k_scatter_max(const float* __restrict__ feat,
                              const int*   __restrict__ src,
                              const int*   __restrict__ dst,
                              unsigned*    __restrict__ agg, int nE) {
    long long t = (long long)blockIdx.x * blockDim.x + threadIdx.x;
    int e = (int)(t >> 4);
    if (e >= nE) return;
    int c4 = ((int)t & 15) << 2;
    int s = src[e], d = dst[e];
    const float4 v = *(const float4*)(feat + (size_t)s * D + c4);
    unsigned* a = agg + (size_t)d * D + c4;
    atomicMax(a + 0, f2mono(v.x));
    atomicMax(a + 1, f2mono(v.y));
    atomicMax(a + 2, f2mono(v.z));
    atomicMax(a + 3, f2mono(v.w));
}

// ---------------------------------------------------------------------------
// Layer 1: h = relu(dec(agg) @ Wl + x @ Wr + b), one 16-row block per CTA,
// 4 wave32s each owning a 16x16 N-tile; 32 V_WMMA_F32_16X16X4_F32 per wave.
__global__ __launch_bounds__(128)
void k_sage_layer1(const unsigned* __restrict__ aggE,
                   const float* __restrict__ x,
                   const float* __restrict__ Wl,
                   const float* __restrict__ Wr,
                   const float* __restrict__ bias,
                   float* __restrict__ h, int nNodes) {
    __shared__ float lW[2 * D * D];        // Wl then Wr (linear: row k * 64 + n)
    __shared__ float lA[2 * 16 * LDA];     // agg tile, x tile (padded stride)
    __shared__ float lB[D];                // bias

    const int tid   = threadIdx.x;
    const int node0 = blockIdx.x * 16;

    // Stage weights: 2 * 4096 floats as float4 (Wl/Wr are row-major [64,64]).
    for (int i4 = tid; i4 < 2 * D * D / 4; i4 += 128) {
        const float4* sp = (i4 < D * D / 4)
                               ? ((const float4*)Wl + i4)
                               : ((const float4*)Wr + (i4 - D * D / 4));
        *(float4*)&lW[i4 * 4] = *sp;
    }
    if (tid < D) lB[tid] = bias[tid];

    // Stage A tiles: 16 rows x 64 cols of decoded agg, and of x.
    for (int i4 = tid; i4 < 2 * 16 * 16; i4 += 128) {   // 512 float4 total
        int which = i4 >> 8;                            // 0 = agg, 1 = x
        int j     = i4 & 255;
        int row   = j >> 4;
        int col4  = (j & 15) << 2;
        float4 v;
        if (node0 + row < nNodes) {
            if (which == 0) {
                uint4 u = *(const uint4*)(aggE + (size_t)(node0 + row) * D + col4);
                v.x = mono2f_or_zero(u.x); v.y = mono2f_or_zero(u.y);
                v.z = mono2f_or_zero(u.z); v.w = mono2f_or_zero(u.w);
            } else {
                v = *(const float4*)(x + (size_t)(node0 + row) * D + col4);
            }
        } else {
            v = make_float4(0.f, 0.f, 0.f, 0.f);
        }
        *(float4*)&lA[which * 16 * LDA + row * LDA + col4] = v;
    }
    __syncthreads();

    const int wave  = tid >> 5;            // 0..3 : which 16-col N tile
    const int lane  = tid & 31;
    const int mn    = lane & 15;           // M for A-frag, N for B/C-frag
    const int khalf = (lane >> 4) << 1;    // lanes 16-31 own K+2 per ISA layout
    const int ncol  = wave * 16 + mn;

    const float* Ag = &lA[0];
    const float* Ax = &lA[16 * LDA];
    const float* Bl = &lW[0];
    const float* Br = &lW[D * D];

    v8f c = {};
    #pragma unroll
    for (int k0 = 0; k0 < D; k0 += 4) {
        const int kb = k0 + khalf;
        v2f aA, aX, bL, bR;
        // A 16x4 f32: lane m, VGPR0=K=kb, VGPR1=K=kb+1 (kb+2 for hi lanes).
        aA.x = Ag[mn * LDA + kb];  aA.y = Ag[mn * LDA + kb + 1];
        aX.x = Ax[mn * LDA + kb];  aX.y = Ax[mn * LDA + kb + 1];
        // B 4x16 f32: row K striped across lanes; VGPR0 rows {0,2}, VGPR1 {1,3}.
        bL.x = Bl[kb * D + ncol];  bL.y = Bl[(kb + 1) * D + ncol];
        bR.x = Br[kb * D + ncol];  bR.y = Br[(kb + 1) * D + ncol];
        c = __builtin_amdgcn_wmma_f32_16x16x4_f32(false, aA, false, bL,
                                                  (short)0, c, false, false);
        c = __builtin_amdgcn_wmma_f32_16x16x4_f32(false, aX, false, bR,
                                                  (short)0, c, false, false);
    }

    // C/D layout: VGPR r -> row r (lanes 0-15) / row r+8 (lanes 16-31).
    const float bb = lB[ncol];
    #pragma unroll
    for (int r = 0; r < 8; ++r) {
        int m   = r + ((lane >> 4) << 3);
        int row = node0 + m;
        if (row < nNodes) {
            float v = c[r] + bb;
            v = v > 0.f ? v : 0.f;          // relu
            h[(size_t)row * D + ncol] = v;
        }
    }
}

// ---------------------------------------------------------------------------
// Layer 2 (d_out = 1): out[i] = dec(agg2[i]) . Wl2 + h[i] . Wr2 + b2
__global__ void k_sage_layer2(const unsigned* __restrict__ aggE,
                              const float* __restrict__ h,
                              const float* __restrict__ Wl2,
                              const float* __restrict__ Wr2,
                              const float* __restrict__ b2,
                              float* __restrict__ out, int nNodes) {
    __shared__ float wl[D], wr[D];
    int tid = threadIdx.x;
    if (tid < D)            wl[tid]     = Wl2[tid];
    else if (tid < 2 * D)   wr[tid - D] = Wr2[tid - D];
    __syncthreads();

    int node = blockIdx.x * blockDim.x + tid;
    if (node >= nNodes) return;

    const unsigned* ap = aggE + (size_t)node * D;
    const float*    hp = h    + (size_t)node * D;
    float acc = b2[0];
    #pragma unroll 8
    for (int k = 0; k < D; ++k)
        acc += mono2f_or_zero(ap[k]) * wl[k] + hp[k] * wr[k];
    out[node] = acc;
}

// ---------------------------------------------------------------------------
extern "C" void kernel_launch(void* const* d_in, const int* in_sizes, int n_in,
                              void* d_out, int out_size, void* d_ws, size_t ws_size,
                              hipStream_t stream) {
    const float* x   = (const float*)d_in[0];
    const int*   ei  = (const int*)  d_in[1];   // [2, E] row-major
    const float* Wl1 = (const float*)d_in[2];
    const float* Wr1 = (const float*)d_in[3];
    const float* b1  = (const float*)d_in[4];
    const float* Wl2 = (const float*)d_in[5];
    const float* Wr2 = (const float*)d_in[6];
    const float* b2  = (const float*)d_in[7];

    const int N = in_sizes[0] / D;              // 50000
    const int E = in_sizes[1] / 2;              // 800000
    const int* src = ei;
    const int* dst = ei + E;

    unsigned* agg = (unsigned*)d_ws;                                   // N*64 u32
    float*    h   = (float*)((char*)d_ws + (size_t)N * D * sizeof(float)); // N*64 f32

    const int nElem = N * D;
    const int initBlocks    = (nElem + 255) / 256;
    const int scatterBlocks = (int)(((long long)E * 16 + 255) / 256);
    const int gemmBlocks    = (N + 15) / 16;
    const int outBlocks     = (N + 255) / 256;

    // Layer 1
    k_init_enc   <<<initBlocks,    256, 0, stream>>>(agg, nElem);
    k_scatter_max<<<scatterBlocks, 256, 0, stream>>>(x, src, dst, agg, E);
    k_sage_layer1<<<gemmBlocks,    128, 0, stream>>>(agg, x, Wl1, Wr1, b1, h, N);
    // Layer 2 (reuse agg buffer)
    k_init_enc   <<<initBlocks,    256, 0, stream>>>(agg, nElem);
    k_scatter_max<<<scatterBlocks, 256, 0, stream>>>(h, src, dst, agg, E);
    k_sage_layer2<<<outBlocks,     256, 0, stream>>>(agg, h, Wl2, Wr2, b2,
                                                     (float*)d_out, N);
}